// AttentionModel_31172872634636
// MI455X (gfx1250) — compile-verified
//
#include <hip/hip_runtime.h>
#include <cstdint>
#include <cstddef>

#define B_ROWS 16384
#define D_DIM  1024
#define NG     64

typedef float v2f __attribute__((ext_vector_type(2)));
typedef float v4f __attribute__((ext_vector_type(4)));
typedef float v8f __attribute__((ext_vector_type(8)));

// ---------------- workspace layout (byte offsets) ----------------
#define WS_PERM   ((size_t)0)                       // int[B]
#define WS_LS     ((size_t)(64*1024))               // int[B]
#define WS_COUNTS ((size_t)(128*1024))              // int[NG]
#define WS_OFFS   (WS_COUNTS + 256)                 // int[NG]
#define WS_FIRST  (WS_OFFS   + 256)                 // int[NG]
#define WS_META   (WS_FIRST  + 256)                 // int/float[4]
#define WS_HID    ((size_t)(256*1024))              // float[NG*D]
#define WS_F      (WS_HID + 256*1024)               // float[NG*D]
#define WS_IMEAN  (WS_F   + 256*1024)               // float[NG*D]
#define WS_T2     ((size_t)(1024*1024))             // float[B*D]   (later: obar)
#define WS_T4     (WS_T2 + (size_t)B_ROWS*D_DIM*4)  // float[B*D]   (later: intra)
#define WS_QKV    (WS_T4 + (size_t)B_ROWS*D_DIM*4)  // float[2B*3D]

// ---------------- sort: histogram / scan / stable rank ----------------
__global__ void k_init(int* counts, int* firstIdx) {
  int t = threadIdx.x;
  if (t < NG) { counts[t] = 0; firstIdx[t] = 0x7fffffff; }
}

__global__ void k_hist(const int* __restrict__ labels, int* counts, int* firstIdx) {
  int i = blockIdx.x * blockDim.x + threadIdx.x;
  if (i < B_ROWS) {
    int g = labels[i];
    atomicAdd(&counts[g], 1);
    atomicMin(&firstIdx[g], i);
  }
}

__global__ void k_scan(const int* __restrict__ counts, int* offs, int* meta) {
  if (threadIdx.x == 0) {
    int run = 0, np = 0;
    for (int g = 0; g < NG; ++g) { offs[g] = run; run += counts[g]; np += (counts[g] > 0); }
    meta[0] = np;
    int dm = np - 1; if (dm < 1) dm = 1;
    ((float*)meta)[1] = (float)dm;
    ((float*)meta)[2] = (np > 1) ? 1.0f : 0.0f;
  }
}

__global__ __launch_bounds__(32)
void k_rank(const int* __restrict__ labels, const int* __restrict__ offs,
            int* __restrict__ perm, int* __restrict__ ls) {
  int g = blockIdx.x;
  int lane = threadIdx.x;
  int run = offs[g];
  for (int c0 = 0; c0 < B_ROWS; c0 += 32) {
    int i = c0 + lane;
    bool m = (labels[i] == g);
    unsigned bal = (unsigned)__ballot(m);       // wave32: low 32 bits
    if (m) {
      int pos = run + __popc(bal & ((1u << lane) - 1u));
      perm[pos] = i;
      ls[pos] = g;
    }
    run += __popc(bal);
  }
}

// ---------------- segment conv (k=2 and k=4), sorted order ----------------
__global__ __launch_bounds__(256)
void k_segconv(const float* __restrict__ x, const int* __restrict__ perm,
               const int* __restrict__ ls,
               const float* __restrict__ w2p, const float* __restrict__ b2p,
               const float* __restrict__ w4p, const float* __restrict__ b4p,
               float* __restrict__ t2, float* __restrict__ t4) {
  int p = blockIdx.x;
  int g = ls[p];
  bool m1 = (p >= 1) && (ls[p - 1] == g);
  bool m2 = (p >= 2) && (ls[p - 2] == g);
  bool q1 = (p + 1 < B_ROWS) && (ls[p + 1] == g);
  const float* x0  = x + (size_t)perm[p] * D_DIM;
  const float* xm1 = m1 ? x + (size_t)perm[p - 1] * D_DIM : x0;
  const float* xm2 = m2 ? x + (size_t)perm[p - 2] * D_DIM : x0;
  const float* xp1 = q1 ? x + (size_t)perm[p + 1] * D_DIM : x0;
  float w20 = w2p[0], w21 = w2p[1], c2 = b2p[0];
  float w40 = w4p[0], w41 = w4p[1], w42 = w4p[2], w43 = w4p[3], c4 = b4p[0];
  float f1 = m1 ? 1.f : 0.f, f2 = m2 ? 1.f : 0.f, fq = q1 ? 1.f : 0.f;
  int d = threadIdx.x * 4;
  v4f v0  = *(const v4f*)(x0  + d);
  v4f vm1 = *(const v4f*)(xm1 + d) * f1;
  v4f vm2 = *(const v4f*)(xm2 + d) * f2;
  v4f vp1 = *(const v4f*)(xp1 + d) * fq;
  v4f r2 = w21 * v0 + w20 * vm1 + c2;
  v4f r4 = w42 * v0 + w41 * vm1 + w40 * vm2 + w43 * vp1 + c4;
  *(v4f*)(t2 + (size_t)p * D_DIM + d) = r2;
  *(v4f*)(t4 + (size_t)p * D_DIM + d) = r4;
}

// ---------------- fp32 WMMA GEMM: C[M,N] = A[M,K] * W[N,K]^T + bias ----------------
// Per-wave 64x64 C tile = 4x4 grid of 16x16 WMMA tiles, K stepped by 4.
// 16 WMMAs per 8 b64 fragment loads -> ~16 FLOP per L2 byte.
template <bool SCATTER>
__global__ __launch_bounds__(256)
void k_wmma_gemm(const float* __restrict__ A, const float* __restrict__ W,
                 const float* __restrict__ bias, float* __restrict__ C,
                 int M, int N, int K,
                 const float* __restrict__ imean, const int* __restrict__ ls,
                 const int* __restrict__ perm) {
  const int wave = (blockIdx.x * blockDim.x + threadIdx.x) >> 5;
  const int lane = threadIdx.x & 31;
  const int tilesN = N >> 6;
  const int tm = wave / tilesN;
  const int tn = wave - tm * tilesN;
  const int m0 = tm * 64;
  const int n0 = tn * 64;
  if (m0 >= M) return;

  const int half = lane >> 4;   // 0: K=0..1  1: K=2..3 (ISA 16x4 f32 A layout)
  const int r16  = lane & 15;
  const int koff = half * 2;

  const float* ap[4];
  const float* bp[4];
  #pragma unroll
  for (int i = 0; i < 4; ++i) {
    ap[i] = A + (size_t)(m0 + i * 16 + r16) * K + koff;
    bp[i] = W + (size_t)(n0 + i * 16 + r16) * K + koff;
  }

  v8f acc[4][4] = {};

  #pragma unroll 2
  for (int k = 0; k < K; k += 4) {
    v2f af[4], bf[4];
    #pragma unroll
    for (int i = 0; i < 4; ++i) af[i] = *(const v2f*)(ap[i] + k);
    #pragma unroll
    for (int j = 0; j < 4; ++j) bf[j] = *(const v2f*)(bp[j] + k);
    #pragma unroll
    for (int i = 0; i < 4; ++i)
      #pragma unroll
      for (int j = 0; j < 4; ++j)
        acc[i][j] = __builtin_amdgcn_wmma_f32_16x16x4_f32(
            false, af[i], false, bf[j], (short)0, acc[i][j], false, false);
  }

  // epilogue: C layout (vgpr r, lane L): M = r + 8*(L>=16), N = L&15
  #pragma unroll
  for (int i = 0; i < 4; ++i) {
    #pragma unroll
    for (int r = 0; r < 8; ++r) {
      int row = m0 + i * 16 + half * 8 + r;
      int pr = 0, g = 0;
      if (SCATTER) { pr = perm[row]; g = ls[row]; }
      #pragma unroll
      for (int j = 0; j < 4; ++j) {
        int col = n0 + j * 16 + r16;
        float v = acc[i][j][r] + bias[col];
        if (SCATTER) {
          v += imean[(size_t)g * D_DIM + col];
          C[(size_t)pr * D_DIM + col] = v;
        } else {
          C[(size_t)row * (size_t)N + col] = v;
        }
      }
    }
  }
}

// ---------------- S=2 attention combine (closed-form softmax) ----------------
__global__ __launch_bounds__(256)
void k_attn(const float* __restrict__ qkv, float* __restrict__ obar) {
  __shared__ float dots[8];   // [head][s][s']
  __shared__ float coef[4];   // [head][j]
  int n = blockIdx.x;
  int tid = threadIdx.x;
  if (tid < 8) dots[tid] = 0.f;
  __syncthreads();
  const float* r0 = qkv + (size_t)n * (3 * D_DIM);
  const float* r1 = qkv + (size_t)(B_ROWS + n) * (3 * D_DIM);
  int d0 = tid * 4;
  int head = d0 >> 9;                  // dh = 512
  float p00 = 0, p01 = 0, p10 = 0, p11 = 0;
  #pragma unroll
  for (int u = 0; u < 4; ++u) {
    int d = d0 + u;
    float q0 = r0[d], q1 = r1[d], k0 = r0[D_DIM + d], k1 = r1[D_DIM + d];
    p00 += q0 * k0; p01 += q0 * k1; p10 += q1 * k0; p11 += q1 * k1;
  }
  atomicAdd(&dots[head * 4 + 0], p00);
  atomicAdd(&dots[head * 4 + 1], p01);
  atomicAdd(&dots[head * 4 + 2], p10);
  atomicAdd(&dots[head * 4 + 3], p11);
  __syncthreads();
  if (tid < 2) {
    const float rs = 0.044194173824159216f;  // 1/sqrt(512)
    float s00 = dots[tid * 4 + 0] * rs, s01 = dots[tid * 4 + 1] * rs;
    float s10 = dots[tid * 4 + 2] * rs, s11 = dots[tid * 4 + 3] * rs;
    float mA = fmaxf(s00, s01), e00 = __expf(s00 - mA), e01 = __expf(s01 - mA);
    float mB = fmaxf(s10, s11), e10 = __expf(s10 - mB), e11 = __expf(s11 - mB);
    float i0 = 1.f / (e00 + e01), i1 = 1.f / (e10 + e11);
    coef[tid * 2 + 0] = 0.5f * (e00 * i0 + e10 * i1);
    coef[tid * 2 + 1] = 0.5f * (e01 * i0 + e11 * i1);
  }
  __syncthreads();
  float c0 = coef[head * 2 + 0], c1 = coef[head * 2 + 1];
  v4f v0 = *(const v4f*)(r0 + 2 * D_DIM + d0);
  v4f v1 = *(const v4f*)(r1 + 2 * D_DIM + d0);
  *(v4f*)(obar + (size_t)n * D_DIM + d0) = c0 * v0 + c1 * v1;
}

// ---------------- inter path (64 group-first rows, tiny GEMMs) ----------------
__global__ __launch_bounds__(256)
void k_inter_hidden(const float* __restrict__ x, const int* __restrict__ firstIdx,
                    const int* __restrict__ counts,
                    const float* __restrict__ iin_w, const float* __restrict__ iin_b,
                    float* __restrict__ hidden) {
  int g = blockIdx.x;
  int fi = (counts[g] > 0) ? firstIdx[g] : 0;
  const float* row = x + (size_t)fi * D_DIM;
  const float* Wv  = iin_w + (size_t)(2 * D_DIM) * D_DIM;
  const float* bv  = iin_b + 2 * D_DIM;
  for (int u = 0; u < 4; ++u) {
    int e = threadIdx.x + 256 * u;
    const float* wr = Wv + (size_t)e * D_DIM;
    float s = bv[e];
    for (int d = 0; d < D_DIM; ++d) s += row[d] * wr[d];
    hidden[g * D_DIM + e] = s;
  }
}

__global__ __launch_bounds__(256)
void k_inter_f(const float* __restrict__ hidden, const int* __restrict__ counts,
               const float* __restrict__ iout_w, const float* __restrict__ iout_b,
               float* __restrict__ f) {
  int g = blockIdx.x;
  bool pres = counts[g] > 0;
  const float* h = hidden + (size_t)g * D_DIM;
  for (int u = 0; u < 4; ++u) {
    int e = threadIdx.x + 256 * u;
    const float* wr = iout_w + (size_t)e * D_DIM;
    float s = iout_b[e];
    for (int d = 0; d < D_DIM; ++d) s += h[d] * wr[d];
    f[g * D_DIM + e] = pres ? s : 0.f;
  }
}

__global__ __launch_bounds__(256)
void k_inter_mean(const float* __restrict__ f, const int* __restrict__ meta,
                  float* __restrict__ imean) {
  float denom = ((const float*)meta)[1];
  float flag  = ((const float*)meta)[2];
  float inv = flag / denom;
  for (int u = 0; u < 4; ++u) {
    int d = threadIdx.x + 256 * u;
    float s = 0.f;
    for (int g = 0; g < NG; ++g) s += f[g * D_DIM + d];
    for (int g = 0; g < NG; ++g) imean[g * D_DIM + d] = (s - f[g * D_DIM + d]) * inv;
  }
}

// ---------------- launcher ----------------
extern "C" void kernel_launch(void* const* d_in, const int* in_sizes, int n_in,
                              void* d_out, int out_size, void* d_ws, size_t ws_size,
                              hipStream_t stream) {
  const float* x      = (const float*)d_in[0];
  const int*   labels = (const int*)d_in[1];
  const float* w2     = (const float*)d_in[3];
  const float* b2     = (const float*)d_in[4];
  const float* w4     = (const float*)d_in[5];
  const float* b4     = (const float*)d_in[6];
  const float* in_w   = (const float*)d_in[7];
  const float* in_b   = (const float*)d_in[8];
  const float* out_w  = (const float*)d_in[9];
  const float* out_b  = (const float*)d_in[10];
  const float* iin_w  = (const float*)d_in[11];
  const float* iin_b  = (const float*)d_in[12];
  const float* iout_w = (const float*)d_in[13];
  const float* iout_b = (const float*)d_in[14];
  const float* lin_w  = (const float*)d_in[15];
  const float* lin_b  = (const float*)d_in[16];

  char* ws = (char*)d_ws;
  int*   perm    = (int*)(ws + WS_PERM);
  int*   ls      = (int*)(ws + WS_LS);
  int*   counts  = (int*)(ws + WS_COUNTS);
  int*   offs    = (int*)(ws + WS_OFFS);
  int*   firstI  = (int*)(ws + WS_FIRST);
  int*   meta    = (int*)(ws + WS_META);
  float* hidden  = (float*)(ws + WS_HID);
  float* fbuf    = (float*)(ws + WS_F);
  float* imean   = (float*)(ws + WS_IMEAN);
  float* t2      = (float*)(ws + WS_T2);   // also [2B,1024] A for GEMM1 (t2|t4 contig)
  float* t4      = (float*)(ws + WS_T4);
  float* qkv     = (float*)(ws + WS_QKV);
  float* obar    = t2;                     // alias after GEMM1
  float* intra   = t4;                     // alias after GEMM2
  float* out     = (float*)d_out;

  // 1) stable counting sort by label
  k_init<<<1, 64, 0, stream>>>(counts, firstI);
  k_hist<<<B_ROWS / 256, 256, 0, stream>>>(labels, counts, firstI);
  k_scan<<<1, 64, 0, stream>>>(counts, offs, meta);
  k_rank<<<NG, 32, 0, stream>>>(labels, offs, perm, ls);

  // 2) segment convolutions -> tok tokens (sorted order)
  k_segconv<<<B_ROWS, 256, 0, stream>>>(x, perm, ls, w2, b2, w4, b4, t2, t4);

  // 3) qkv = [t2;t4] @ intra_in_w^T + b   (2B x 3072 x K=1024)
  {
    int M = 2 * B_ROWS, N = 3 * D_DIM, K = D_DIM;
    int blocks = ((M / 64) * (N / 64)) / 8;
    k_wmma_gemm<false><<<blocks, 256, 0, stream>>>(t2, in_w, in_b, qkv, M, N, K,
                                                   nullptr, nullptr, nullptr);
  }

  // 4) S=2 attention, closed form; mean over S folded in -> obar
  k_attn<<<B_ROWS, 256, 0, stream>>>(qkv, obar);

  // 5) inter path (tiny)
  k_inter_hidden<<<NG, 256, 0, stream>>>(x, firstI, counts, iin_w, iin_b, hidden);
  k_inter_f<<<NG, 256, 0, stream>>>(hidden, counts, iout_w, iout_b, fbuf);
  k_inter_mean<<<1, 256, 0, stream>>>(fbuf, meta, imean);

  // 6) intra = obar @ intra_out_w^T + b
  {
    int M = B_ROWS, N = D_DIM, K = D_DIM;
    int blocks = ((M / 64) * (N / 64)) / 8;
    k_wmma_gemm<false><<<blocks, 256, 0, stream>>>(obar, out_w, out_b, intra, M, N, K,
                                                   nullptr, nullptr, nullptr);
  }

  // 7) out[perm[row]] = intra @ lin_w^T + lin_b + inter_mean[ls[row]]
  {
    int M = B_ROWS, N = D_DIM, K = D_DIM;
    int blocks = ((M / 64) * (N / 64)) / 8;
    k_wmma_gemm<true><<<blocks, 256, 0, stream>>>(intra, lin_w, lin_b, out, M, N, K,
                                                  imean, ls, perm);
  }
}